// Gemma3nAudioRelativePositionEmbedding_88441966559355
// MI455X (gfx1250) — compile-verified
//
#include <hip/hip_runtime.h>
#include <math.h>

typedef __attribute__((ext_vector_type(2))) float v2f;
typedef __attribute__((ext_vector_type(8))) float v8f;

#define HIDDEN   1536
#define HEADS    8
#define HEAD_DIM 192
#define NBATCH   16
#define NBLK     128
#define QBS      12
#define KCS      24
#define FSPAN    13

// ---------------- kernel 1: sin_emb[13][1536] = timing_signal @ W ----------------
// grid = 6 (column slices of 256), block = 256. Each block reads its W column
// slice exactly once (W touched once in total, 9.4 MB); timing-signal tiles are
// built in LDS so transcendentals are amortized (13 values per thread per tile).
__global__ void sinemb_kernel(const float* __restrict__ W, float* __restrict__ sin_emb) {
    __shared__ float ts[FSPAN * 256];
    const int tid = threadIdx.x;
    const int j = blockIdx.x * 256 + tid;
    const float log_inc = logf(10000.0f) / 767.0f;  // HIDDEN/2 - 1 = 767

    float acc[FSPAN];
#pragma unroll
    for (int f = 0; f < FSPAN; ++f) acc[f] = 0.0f;

    for (int i0 = 0; i0 < HIDDEN; i0 += 256) {
        __syncthreads();
        for (int idx = tid; idx < FSPAN * 256; idx += 256) {
            const int f  = idx >> 8;
            const int ii = idx & 255;
            const int i  = i0 + ii;
            const float pos = (float)(12 - f);       // positions 12..0 descending
            const bool  is_sin = (i < HIDDEN / 2);
            const float invv = expf(-log_inc * (float)(is_sin ? i : i - HIDDEN / 2));
            const float arg  = pos * invv;
            ts[idx] = is_sin ? sinf(arg) : cosf(arg);
        }
        __syncthreads();
        for (int ii = 0; ii < 256; ++ii) {
            const float w = W[(size_t)(i0 + ii) * HIDDEN + j];
#pragma unroll
            for (int f = 0; f < FSPAN; ++f) acc[f] += ts[f * 256 + ii] * w;
        }
    }
#pragma unroll
    for (int f = 0; f < FSPAN; ++f) sin_emb[(size_t)f * HIDDEN + j] = acc[f];
}

// ---------------- kernel 2: shifted[b][h][t][s] (only nb==0 contributes) ----------------
// flat out idx = t*24+s; source = term_bd[b,h,0, idx/25, idx%25], zero if idx%25 >= 13.
__global__ void shifted_kernel(const float* __restrict__ Q,
                               const float* __restrict__ sin_emb,
                               float* __restrict__ shifted) {
    const int bh = blockIdx.x;            // 0..127  (b*8 + h)
    const int b  = bh / HEADS;
    const int h  = bh % HEADS;
    const int idx = threadIdx.x;          // 0..287 output flat index
    const int tp = idx / (KCS + 1);       // /25
    const int fp = idx % (KCS + 1);
    float acc = 0.0f;
    if (fp < FSPAN) {
        const float* q = Q + ((size_t)(b * NBLK) * QBS + tp) * HIDDEN + h * HEAD_DIM;
        const float* e = sin_emb + (size_t)fp * HIDDEN + h * HEAD_DIM;
#pragma unroll 8
        for (int d = 0; d < HEAD_DIM; ++d) acc += q[d] * e[d];
    }
    shifted[(size_t)bh * QBS * KCS + idx] = acc;
}

// ---------------- kernel 3: main batched QK^T with fp32 WMMA ----------------
// One workgroup = (b, nb, head-pair). 64 threads = 2 waves, one head each.
// LDS stages Q (12x384) and K (24x384) slices with coalesced b128 loads; row
// stride padded to 388 words (388 % 64 == 4) so the per-lane ds_load_b64
// fragment reads are bank-conflict-free.
#define HPG   2                       // heads per workgroup
#define COLS  (HPG * HEAD_DIM)        // 384
#define STR   (COLS + 4)              // 388 padded row stride (floats)
#define LDS_BYTES ((QBS + KCS) * STR * 4)   // 55,872 B

__global__ void __launch_bounds__(64) attn_wmma_kernel(const float* __restrict__ Q,
                                                       const float* __restrict__ K,
                                                       const float* __restrict__ shifted,
                                                       float* __restrict__ out) {
    extern __shared__ float lds[];
    float* Qs = lds;                  // [QBS][STR]
    float* Ks = lds + QBS * STR;      // [KCS][STR]

    const int tid = threadIdx.x;
    const int blk = blockIdx.x;
    const int hq  = blk & 3;          // head-pair index (0..3)
    const int bn  = blk >> 2;         // b*128 + nb
    const int b   = bn / NBLK;
    const int nb  = bn % NBLK;

    // ---- cooperative staging (float4, fully coalesced) ----
    const float4* Qg = (const float4*)(Q + (size_t)bn * QBS * HIDDEN + hq * COLS);
    for (int i = tid; i < QBS * (COLS / 4); i += 64) {
        const int r = i / (COLS / 4), c = i % (COLS / 4);
        *(float4*)(Qs + r * STR + 4 * c) = Qg[r * (HIDDEN / 4) + c];
    }
    const float4* Kg = (const float4*)(K + (size_t)bn * KCS * HIDDEN + hq * COLS);
    for (int i = tid; i < KCS * (COLS / 4); i += 64) {
        const int r = i / (COLS / 4), c = i % (COLS / 4);
        *(float4*)(Ks + r * STR + 4 * c) = Kg[r * (HIDDEN / 4) + c];
    }
    __syncthreads();

    // ---- per-wave 12x24x192 GEMM via V_WMMA_F32_16X16X4_F32 ----
    const int hl   = tid >> 5;        // local head (wave id)
    const int lane = tid & 31;
    const int half = lane >> 4;       // 0: k={0,1}, 1: k={2,3}
    const int l15  = lane & 15;
    const int h    = hq * HPG + hl;   // global head

    // A rows: M = l15 (clamp garbage rows 12..15 to stay in-bounds; never stored)
    const int tA = (l15 < QBS) ? l15 : (QBS - 1);
    // B tile1 cols: s = 16 + l15 (clamp 24..31)
    const int s1 = (l15 + 16 < KCS) ? (l15 + 16) : (KCS - 1);

    const float* aRow  = Qs + tA  * STR + hl * HEAD_DIM + 2 * half;
    const float* b0Row = Ks + l15 * STR + hl * HEAD_DIM + 2 * half;
    const float* b1Row = Ks + s1  * STR + hl * HEAD_DIM + 2 * half;

    v8f acc0 = {};
    v8f acc1 = {};
#pragma unroll 4
    for (int kk = 0; kk < HEAD_DIM / 4; ++kk) {
        const v2f a  = *(const v2f*)(aRow  + 4 * kk);
        const v2f b0 = *(const v2f*)(b0Row + 4 * kk);
        const v2f b1 = *(const v2f*)(b1Row + 4 * kk);
        acc0 = __builtin_amdgcn_wmma_f32_16x16x4_f32(false, a, false, b0,
                                                     (short)0, acc0, false, false);
        acc1 = __builtin_amdgcn_wmma_f32_16x16x4_f32(false, a, false, b1,
                                                     (short)0, acc1, false, false);
    }

    // ---- epilogue: add shifted (L2-resident, 147 KB unique) and store ----
    const float* sh = shifted + (size_t)(b * HEADS + h) * QBS * KCS;
    float* og = out + ((size_t)(b * HEADS + h) * NBLK + nb) * QBS * KCS;
#pragma unroll
    for (int i = 0; i < 8; ++i) {
        const int M = i + 8 * half;               // D layout: VGPR i -> M=i / M=i+8
        if (M < QBS) {
            const int s0 = l15;                   // tile0: s = 0..15
            og[M * KCS + s0] = acc0[i] + sh[M * KCS + s0];
            if (l15 < KCS - 16) {                 // tile1: s = 16..23
                const int s = 16 + l15;
                og[M * KCS + s] = acc1[i] + sh[M * KCS + s];
            }
        }
    }
}

// ---------------- host ----------------
extern "C" void kernel_launch(void* const* d_in, const int* in_sizes, int n_in,
                              void* d_out, int out_size, void* d_ws, size_t ws_size,
                              hipStream_t stream) {
    (void)in_sizes; (void)n_in; (void)out_size; (void)ws_size;
    const float* Q = (const float*)d_in[0];   // [16,128,12,8,192]
    const float* K = (const float*)d_in[1];   // [16,128,24,8,192]
    const float* W = (const float*)d_in[2];   // [1536,1536]
    float* out = (float*)d_out;               // [16,8,128,12,24]

    float* sin_emb = (float*)d_ws;                         // 13*1536 floats
    float* shifted = sin_emb + FSPAN * HIDDEN;             // 16*8*12*24 floats

    sinemb_kernel<<<HIDDEN / 256, 256, 0, stream>>>(W, sin_emb);
    shifted_kernel<<<NBATCH * HEADS, QBS * KCS, 0, stream>>>(Q, sin_emb, shifted);
    attn_wmma_kernel<<<NBATCH * NBLK * 4, 64, LDS_BYTES, stream>>>(Q, K, shifted, out);
}